// DepthWise_62989990363478
// MI455X (gfx1250) — compile-verified
//
#include <hip/hip_runtime.h>
#include <math.h>

typedef __attribute__((ext_vector_type(2))) float v2f;
typedef __attribute__((ext_vector_type(8))) float v8f;

#define C_IN  128
#define GCH   256
#define C_OUT 128
#define KNB   27

__device__ __forceinline__ v8f wmma_f32(v2f a, v2f b, v8f c) {
    // D = A(16x4,f32) * B(4x16,f32) + C(16x16,f32)
    return __builtin_amdgcn_wmma_f32_16x16x4_f32(
        /*neg_a=*/false, a, /*neg_b=*/false, b,
        /*c_mod=*/(short)0, c, /*reuse_a=*/false, /*reuse_b=*/false);
}

// ---------------------------------------------------------------------------
// Kernel 0: zero the GRN sum-of-squares accumulator (128 floats)
// ---------------------------------------------------------------------------
__global__ __launch_bounds__(C_OUT) void k_zero(float* __restrict__ p) {
    p[threadIdx.x] = 0.0f;
}

// ---------------------------------------------------------------------------
// Kernel 1: h = LayerNorm(x @ W1 + b1) * ln_g + ln_b     [N,128] -> [N,256]
// 2 waves / block, each wave computes a 16-row x 256-col tile with 16
// v8f WMMA accumulators, K=128 stepped by 4.
// ---------------------------------------------------------------------------
__global__ __launch_bounds__(64) void k_gemm1_ln(
    const float* __restrict__ x, const float* __restrict__ W1,
    const float* __restrict__ b1, const float* __restrict__ lng,
    const float* __restrict__ lnb, float* __restrict__ h, int n)
{
    __shared__ float sh[2][16][257];     // padded to dodge bank conflicts
    __shared__ float mu_s[2][16];
    __shared__ float rs_s[2][16];

    const int lane = threadIdx.x & 31;
    const int wave = threadIdx.x >> 5;
    const int ll   = lane & 15;          // M (rows 0..15) / N (cols 0..15)
    const int pr   = lane >> 4;          // K-pair select (0: K=0,1  1: K=2,3)
    const int r0   = (blockIdx.x * 2 + wave) * 16;

    int ar = r0 + ll;
    if (ar >= n) ar = n - 1;             // clamp loads; stores guarded below
    const float* __restrict__ xrow = x + (size_t)ar * C_IN;

    v8f acc[16];
    #pragma unroll
    for (int t = 0; t < 16; ++t) { v8f z = {}; acc[t] = z; }

    #pragma unroll 1
    for (int k0 = 0; k0 < C_IN; k0 += 4) {
        v2f a;
        a.x = xrow[k0 + pr * 2 + 0];
        a.y = xrow[k0 + pr * 2 + 1];
        const float* __restrict__ wrow = W1 + (size_t)(k0 + pr * 2) * GCH;
        #pragma unroll
        for (int t = 0; t < 16; ++t) {
            v2f b;
            b.x = wrow[t * 16 + ll];
            b.y = wrow[GCH + t * 16 + ll];
            acc[t] = wmma_f32(a, b, acc[t]);
        }
    }

    // Spill C tiles (+b1) to LDS: VGPR r holds row (r + pr*8), col = t*16+ll
    #pragma unroll
    for (int t = 0; t < 16; ++t) {
        const int   col = t * 16 + ll;
        const float bv  = b1[col];
        #pragma unroll
        for (int r = 0; r < 8; ++r)
            sh[wave][r + pr * 8][col] = acc[t][r] + bv;
    }
    __syncthreads();

    // Per-row mean / rstd over 256 channels (lanes 0..15 own one row each)
    if (lane < 16) {
        float s = 0.0f, q = 0.0f;
        #pragma unroll 4
        for (int c = 0; c < GCH; ++c) {
            float v = sh[wave][lane][c];
            s += v; q += v * v;
        }
        float mu  = s * (1.0f / GCH);
        float var = q * (1.0f / GCH) - mu * mu;
        mu_s[wave][lane] = mu;
        rs_s[wave][lane] = rsqrtf(var + 1e-6f);
    }
    __syncthreads();

    // Normalize, affine, store h
    for (int e = lane; e < 16 * GCH; e += 32) {
        int row = e >> 8;
        int col = e & (GCH - 1);
        int gr  = r0 + row;
        if (gr < n) {
            float v = (sh[wave][row][col] - mu_s[wave][row]) * rs_s[wave][row];
            h[(size_t)gr * GCH + col] = v * lng[col] + lnb[col];
        }
    }
}

// ---------------------------------------------------------------------------
// Kernel 2: per 16-point tile:
//   dw[p][c] = sum_k h[nbr[p][k]][c] * dw_w[k][c] + dw_b[c]   (gather conv)
//   y = GELU(dw @ W2 + b2)  -> d_out, plus per-channel sum(y^2) for GRN
// 256 threads (8 waves). Phase 1: gather into LDS (row stride 260 so the
// WMMA A-fragment LDS reads land on distinct banks). Phase 2: each wave
// does one 16x16 output tile with K=256 WMMA loop.
// ---------------------------------------------------------------------------
__global__ __launch_bounds__(256) void k_dw_gemm2_gelu(
    const float* __restrict__ h, const int* __restrict__ nbr,
    const float* __restrict__ dww, const float* __restrict__ dwb,
    const float* __restrict__ W2, const float* __restrict__ b2,
    float* __restrict__ y, float* __restrict__ gss, int n)
{
    __shared__ float dsh[16 * 260];
    __shared__ float ss[C_OUT];

    const int tid  = threadIdx.x;
    const int lane = tid & 31;
    const int wave = tid >> 5;
    const int ll   = lane & 15;
    const int pr   = lane >> 4;
    const int p0   = blockIdx.x * 16;

    if (tid < C_OUT) ss[tid] = 0.0f;

    // ---- Phase 1: depthwise gather. thread -> (point p, 16-channel chunk)
    {
        const int p  = tid >> 4;
        const int c0 = (tid & 15) * 16;
        int pt = p0 + p;
        if (pt >= n) pt = n - 1;
        const int* __restrict__ nrow = nbr + (size_t)pt * KNB;

        float a[16];
        #pragma unroll
        for (int j = 0; j < 16; ++j) a[j] = 0.0f;

        #pragma unroll 1
        for (int k = 0; k < KNB; ++k) {
            const int idx = nrow[k];
            const float* __restrict__ hp = h   + (size_t)idx * GCH + c0;
            const float* __restrict__ wp = dww + (size_t)k   * GCH + c0;
            #pragma unroll
            for (int j = 0; j < 16; ++j)
                a[j] = fmaf(hp[j], wp[j], a[j]);
        }
        #pragma unroll
        for (int j = 0; j < 16; ++j)
            dsh[p * 260 + c0 + j] = a[j] + dwb[c0 + j];
    }
    __syncthreads();

    // ---- Phase 2: GEMM2 tile (wave -> 16 output columns), K=256
    const int n0 = wave * 16;
    v8f c = {};
    #pragma unroll 1
    for (int k0 = 0; k0 < GCH; k0 += 4) {
        v2f av, bv;
        av.x = dsh[ll * 260 + k0 + pr * 2 + 0];
        av.y = dsh[ll * 260 + k0 + pr * 2 + 1];
        const float* __restrict__ w2r = W2 + (size_t)(k0 + pr * 2) * C_OUT + n0 + ll;
        bv.x = w2r[0];
        bv.y = w2r[C_OUT];
        c = wmma_f32(av, bv, c);
    }

    // bias + exact-erf GELU + store + per-channel sum of squares
    float       local = 0.0f;
    const int   col   = n0 + ll;
    const float b2v   = b2[col];
    #pragma unroll
    for (int r = 0; r < 8; ++r) {
        int gr = p0 + r + pr * 8;
        if (gr < n) {
            float v = c[r] + b2v;
            float g = 0.5f * v * (1.0f + erff(v * 0.70710678118654752f));
            y[(size_t)gr * C_OUT + col] = g;
            local += g * g;
        }
    }
    atomicAdd(&ss[col], local);
    __syncthreads();
    if (tid < C_OUT) atomicAdd(&gss[tid], ss[tid]);
}

// ---------------------------------------------------------------------------
// Kernel 3: GRN scale. Gx[c]=sqrt(sumsq[c]); Nx=Gx/(mean(Gx)+eps);
// scale[c] = grn_g[c]*Nx + 1   (since h' = h*(grn_g*Nx+1) + grn_b)
// ---------------------------------------------------------------------------
__global__ __launch_bounds__(C_OUT) void k_grn_scale(
    const float* __restrict__ gss, const float* __restrict__ grng,
    float* __restrict__ scale)
{
    __shared__ float red[C_OUT];
    const int c = threadIdx.x;
    float g = sqrtf(gss[c]);
    red[c] = g;
    __syncthreads();
    #pragma unroll
    for (int s = C_OUT / 2; s > 0; s >>= 1) {
        if (c < s) red[c] += red[c + s];
        __syncthreads();
    }
    float mean = red[0] * (1.0f / C_OUT);
    float nx   = g / (mean + 1e-6f);
    scale[c]   = grng[c] * nx + 1.0f;
}

// ---------------------------------------------------------------------------
// Kernel 4: out = out*scale[c] + grn_b[c] + x   (float4, in place on d_out)
// ---------------------------------------------------------------------------
__global__ __launch_bounds__(256) void k_grn_apply(
    const float* __restrict__ x, const float* __restrict__ scale,
    const float* __restrict__ grnb, float* __restrict__ out, long long total4)
{
    long long i = (long long)blockIdx.x * blockDim.x + threadIdx.x;
    if (i >= total4) return;
    const int c0 = ((int)(i & (C_OUT / 4 - 1))) * 4;
    float4 o  = ((const float4*)out)[i];
    float4 xv = ((const float4*)x)[i];
    float4 sc = *(const float4*)(scale + c0);
    float4 gb = *(const float4*)(grnb + c0);
    o.x = fmaf(o.x, sc.x, gb.x + xv.x);
    o.y = fmaf(o.y, sc.y, gb.y + xv.y);
    o.z = fmaf(o.z, sc.z, gb.z + xv.z);
    o.w = fmaf(o.w, sc.w, gb.w + xv.w);
    ((float4*)out)[i] = o;
}

// ---------------------------------------------------------------------------
extern "C" void kernel_launch(void* const* d_in, const int* in_sizes, int n_in,
                              void* d_out, int out_size, void* d_ws, size_t ws_size,
                              hipStream_t stream)
{
    const float* x    = (const float*)d_in[0];
    const int*   nbr  = (const int*)  d_in[1];
    const float* W1   = (const float*)d_in[2];
    const float* b1   = (const float*)d_in[3];
    const float* lng  = (const float*)d_in[4];
    const float* lnb  = (const float*)d_in[5];
    const float* dww  = (const float*)d_in[6];
    const float* dwb  = (const float*)d_in[7];
    const float* W2   = (const float*)d_in[8];
    const float* b2   = (const float*)d_in[9];
    const float* grng = (const float*)d_in[10];
    const float* grnb = (const float*)d_in[11];
    float* out = (float*)d_out;

    const int n = in_sizes[0] / C_IN;

    // Workspace: h [n,256] fp32, then sumsq[128], scale[128]
    float* h     = (float*)d_ws;
    float* gss   = h + (size_t)n * GCH;
    float* scale = gss + C_OUT;

    k_zero<<<1, C_OUT, 0, stream>>>(gss);

    const int blkA = (n + 31) / 32;         // 2 waves x 16 rows per block
    k_gemm1_ln<<<blkA, 64, 0, stream>>>(x, W1, b1, lng, lnb, h, n);

    const int blkB = (n + 15) / 16;         // 16 points per block
    k_dw_gemm2_gelu<<<blkB, 256, 0, stream>>>(h, nbr, dww, dwb, W2, b2,
                                              out, gss, n);

    k_grn_scale<<<1, C_OUT, 0, stream>>>(gss, grng, scale);

    const long long total4 = (long long)out_size / 4;
    const int blkD = (int)((total4 + 255) / 256);
    k_grn_apply<<<blkD, 256, 0, stream>>>(x, scale, grnb, out, total4);
}